// Voxelization_74921409511808
// MI455X (gfx1250) — compile-verified
//
#include <hip/hip_runtime.h>
#include <hip/hip_bf16.h>
#include <stdint.h>

// ---- problem constants (match reference exactly) ----
#define GX 1408
#define GY 1600
#define GZ 40
#define GYZ 64000            // GY*GZ
#define NWORDS 2816000       // (GX*GYZ)/32 = 90,112,000/32
#define MAX_VOX 120000
#define MAX_PTS 10
#define NPTS 1000000
#define WORDS_PER_BLOCK 1024 // 256 threads x 4 words (b128 per lane)
#define NSCAN_BLOCKS (NWORDS / WORDS_PER_BLOCK)  // 2750
#define EMPTY_IDX 0x7FFFFFFFu

// CDNA5 async global->LDS path (gfx1250). Guarded so the file still compiles
// cleanly on the host pass / if the builtin is named differently.
#if defined(__AMDGCN__) && __has_builtin(__builtin_amdgcn_global_load_async_to_lds_b128) && __has_builtin(__builtin_amdgcn_s_wait_asynccnt)
#define USE_ASYNC_LDS 1
typedef int v4i __attribute__((ext_vector_type(4)));
typedef __attribute__((address_space(1))) v4i* gptr_v4i;
typedef __attribute__((address_space(3))) v4i* lptr_v4i;
#else
#define USE_ASYNC_LDS 0
#endif

// ---- helpers ----
static __device__ __forceinline__ int coords_lin(float x, float y, float z, bool& valid) {
  // Reference: floor((pts - lo) / vs) in fp32, with true division (not rcp).
  int cx = (int)floorf((x - 0.0f)  / 0.05f);
  int cy = (int)floorf((y + 40.0f) / 0.05f);
  int cz = (int)floorf((z + 3.0f)  / 0.1f);
  valid = (cx >= 0) & (cx < GX) & (cy >= 0) & (cy < GY) & (cz >= 0) & (cz < GZ);
  return cx * GYZ + cy * GZ + cz;
}

// ---- fills ----
__global__ void k_fill_u32(uint32_t* __restrict__ p, size_t n, uint32_t v) {
  size_t i = (size_t)blockIdx.x * blockDim.x + threadIdx.x;
  size_t stride = (size_t)gridDim.x * blockDim.x;
  for (; i < n; i += stride) p[i] = v;
}
__global__ void k_fill_f32(float* __restrict__ p, size_t n, float v) {
  size_t i = (size_t)blockIdx.x * blockDim.x + threadIdx.x;
  size_t stride = (size_t)gridDim.x * blockDim.x;
  for (; i < n; i += stride) p[i] = v;
}

// ---- pass 1: mark occupied voxels in bitmap ----
__global__ void k_mark(const float* __restrict__ pts, uint32_t* __restrict__ bitmap) {
  int b = blockIdx.y;
  int i = blockIdx.x * blockDim.x + threadIdx.x;
  if (i >= NPTS) return;
  const float* p = pts + ((size_t)b * NPTS + i) * 3;
  __builtin_prefetch(p + 3 * 4096, 0, 0);   // global_prefetch_b8 (gfx1250)
  bool valid;
  int lin = coords_lin(p[0], p[1], p[2], valid);
  if (!valid) return;
  atomicOr(&bitmap[(size_t)b * NWORDS + (lin >> 5)], 1u << (lin & 31));
}

// ---- pass 2: per-1024-word popcount block sums ----
__global__ void k_block_reduce(const uint32_t* __restrict__ bitmap,
                               uint32_t* __restrict__ blockSums) {
  int b = blockIdx.y, blk = blockIdx.x, t = threadIdx.x;
  const uint32_t* bm = bitmap + (size_t)b * NWORDS + (size_t)blk * WORDS_PER_BLOCK;
  uint32_t w0, w1, w2, w3;
#if USE_ASYNC_LDS
  __shared__ uint32_t lds_w[WORDS_PER_BLOCK];
  __builtin_amdgcn_global_load_async_to_lds_b128(
      (gptr_v4i)(bm + (size_t)t * 4),
      (lptr_v4i)&lds_w[t * 4], 0, 0);
  __builtin_amdgcn_s_wait_asynccnt(0);
  __syncthreads();
  w0 = lds_w[t * 4 + 0]; w1 = lds_w[t * 4 + 1];
  w2 = lds_w[t * 4 + 2]; w3 = lds_w[t * 4 + 3];
#else
  const uint4 w = *(const uint4*)(bm + (size_t)t * 4);
  w0 = w.x; w1 = w.y; w2 = w.z; w3 = w.w;
#endif
  uint32_t s = __popc(w0) + __popc(w1) + __popc(w2) + __popc(w3);
  __shared__ uint32_t red[256];
  red[t] = s; __syncthreads();
  for (int off = 128; off > 0; off >>= 1) {
    if (t < off) red[t] += red[t + off];
    __syncthreads();
  }
  if (t == 0) blockSums[(size_t)b * NSCAN_BLOCKS + blk] = red[0];
}

// ---- pass 3: exclusive scan of the 2750 block sums (one block per batch) ----
__global__ void k_scan_blocksums(uint32_t* __restrict__ blockSums) {
  int b = blockIdx.x, t = threadIdx.x;
  uint32_t* bs = blockSums + (size_t)b * NSCAN_BLOCKS;
  __shared__ uint32_t sh[256];
  __shared__ uint32_t carry_s;
  if (t == 0) carry_s = 0;
  __syncthreads();
  for (int base = 0; base < NSCAN_BLOCKS; base += 256) {
    int i = base + t;
    uint32_t v = (i < NSCAN_BLOCKS) ? bs[i] : 0u;
    uint32_t carry = carry_s;
    sh[t] = v; __syncthreads();
    for (int off = 1; off < 256; off <<= 1) {
      uint32_t u = (t >= off) ? sh[t - off] : 0u;
      __syncthreads();
      sh[t] += u;
      __syncthreads();
    }
    uint32_t incl = sh[t];
    if (i < NSCAN_BLOCKS) bs[i] = carry + incl - v;   // exclusive
    __syncthreads();
    if (t == 255) carry_s = carry + incl;
    __syncthreads();
  }
}

// ---- pass 4: per-word exclusive prefix + emit vc/mask for ranks < MAX_VOX ----
__global__ void k_scan_emit(const uint32_t* __restrict__ bitmap,
                            const uint32_t* __restrict__ blockSums,
                            uint32_t* __restrict__ prefix,
                            float* __restrict__ outVC,
                            float* __restrict__ outMask) {
  int b = blockIdx.y, blk = blockIdx.x, t = threadIdx.x;
  size_t wbase = (size_t)blk * WORDS_PER_BLOCK + (size_t)t * 4;
  const uint32_t* bm = bitmap + (size_t)b * NWORDS;
  const uint4 w = *(const uint4*)(bm + wbase);
  uint32_t pc[4] = { (uint32_t)__popc(w.x), (uint32_t)__popc(w.y),
                     (uint32_t)__popc(w.z), (uint32_t)__popc(w.w) };
  uint32_t tsum = pc[0] + pc[1] + pc[2] + pc[3];
  __shared__ uint32_t sh[256];
  sh[t] = tsum; __syncthreads();
  for (int off = 1; off < 256; off <<= 1) {
    uint32_t u = (t >= off) ? sh[t - off] : 0u;
    __syncthreads();
    sh[t] += u;
    __syncthreads();
  }
  uint32_t base = blockSums[(size_t)b * NSCAN_BLOCKS + blk] + (sh[t] - tsum);
  uint32_t* pr = prefix + (size_t)b * NWORDS + wbase;
  pr[0] = base;
  pr[1] = base + pc[0];
  pr[2] = base + pc[0] + pc[1];
  pr[3] = base + pc[0] + pc[1] + pc[2];

  float* vcB = outVC + (size_t)b * MAX_VOX * 3;
  float* mkB = outMask + (size_t)b * MAX_VOX;
  uint32_t words[4] = { w.x, w.y, w.z, w.w };
  uint32_t r = base;
  for (int j = 0; j < 4; ++j) {
    uint32_t bits = words[j];
    uint32_t rr = r;
    while (bits) {
      if (rr >= MAX_VOX) break;
      int bp = __builtin_ctz(bits);
      int lin = (int)((wbase + (size_t)j) * 32) + bp;
      int xq = lin / GYZ; int rem = lin - xq * GYZ;
      int yq = rem / GZ;  int zq = rem - yq * GZ;
      size_t o = (size_t)rr * 3;
      vcB[o + 0] = (float)xq;
      vcB[o + 1] = (float)yq;
      vcB[o + 2] = (float)zq;
      mkB[rr] = 1.0f;
      bits &= bits - 1; rr++;
    }
    r += pc[j];
  }
}

// ---- pass 5: deterministic index-ordered slot insertion (atomicMin trickle) ----
__global__ void k_scatter(const float* __restrict__ pts,
                          const uint32_t* __restrict__ bitmap,
                          const uint32_t* __restrict__ prefix,
                          uint32_t* __restrict__ slotIdx) {
  int b = blockIdx.y;
  int i = blockIdx.x * blockDim.x + threadIdx.x;
  if (i >= NPTS) return;
  const float* p = pts + ((size_t)b * NPTS + i) * 3;
  __builtin_prefetch(p + 3 * 4096, 0, 0);
  bool valid;
  int lin = coords_lin(p[0], p[1], p[2], valid);
  if (!valid) return;
  int wi = lin >> 5, bi = lin & 31;
  uint32_t word = bitmap[(size_t)b * NWORDS + wi];
  uint32_t rank = prefix[(size_t)b * NWORDS + wi] + __popc(word & ((1u << bi) - 1u));
  if (rank >= MAX_VOX) return;
  uint32_t* s = slotIdx + ((size_t)b * MAX_VOX + rank) * MAX_PTS;
  uint32_t v = (uint32_t)i;
  // Keeps the MAX_PTS smallest point indices, sorted -- identical to the
  // reference's stable-argsort slot assignment, independent of atomic order.
  #pragma unroll
  for (int k = 0; k < MAX_PTS; ++k) {
    uint32_t old = atomicMin(&s[k], v);
    if (old == EMPTY_IDX) break;   // landed in an empty slot, nothing displaced
    if (old > v) v = old;          // carry displaced larger index downstream
  }
}

// ---- pass 6: dense voxels write (zeros for empty slots) ----
__global__ void k_write_voxels(const float* __restrict__ pts,
                               const uint32_t* __restrict__ slotIdx,
                               float* __restrict__ outVox,
                               int batches) {
  size_t total = (size_t)batches * MAX_VOX * MAX_PTS;
  size_t i = (size_t)blockIdx.x * blockDim.x + threadIdx.x;
  if (i >= total) return;
  int b = (int)(i / ((size_t)MAX_VOX * MAX_PTS));
  uint32_t idx = slotIdx[i];
  float x = 0.f, y = 0.f, z = 0.f;
  if (idx != EMPTY_IDX) {
    const float* p = pts + ((size_t)b * NPTS + idx) * 3;
    x = p[0]; y = p[1]; z = p[2];
  }
  float* o = outVox + i * 3;
  o[0] = x; o[1] = y; o[2] = z;
}

extern "C" void kernel_launch(void* const* d_in, const int* in_sizes, int n_in,
                              void* d_out, int out_size, void* d_ws, size_t ws_size,
                              hipStream_t stream) {
  const float* pts = (const float*)d_in[0];
  int B = in_sizes[0] / (NPTS * 3);
  if (B < 1) B = 1;

  float* out     = (float*)d_out;
  float* outVox  = out;                                             // (B,120000,10,3)
  float* outVC   = out + (size_t)B * MAX_VOX * MAX_PTS * 3;         // (B,120000,3) as float
  float* outMask = outVC + (size_t)B * MAX_VOX * 3;                 // (B,120000) as 0/1 float

  uint32_t* ws        = (uint32_t*)d_ws;
  uint32_t* bitmap    = ws;                                         // B*NWORDS
  uint32_t* prefix    = bitmap + (size_t)B * NWORDS;                // B*NWORDS
  uint32_t* blockSums = prefix + (size_t)B * NWORDS;                // B*NSCAN_BLOCKS
  uint32_t* slotIdx   = blockSums + (size_t)B * NSCAN_BLOCKS;       // B*MAX_VOX*MAX_PTS

  const int T = 256;

  k_fill_u32<<<2048, T, 0, stream>>>(bitmap, (size_t)B * NWORDS, 0u);
  k_fill_u32<<<2048, T, 0, stream>>>(slotIdx, (size_t)B * MAX_VOX * MAX_PTS, EMPTY_IDX);
  // vc (3 floats) and mask (1 float) regions are contiguous: clear both at once
  k_fill_f32<<<1024, T, 0, stream>>>(outVC, (size_t)B * MAX_VOX * 4, 0.0f);

  dim3 gP((NPTS + T - 1) / T, B);
  k_mark<<<gP, T, 0, stream>>>(pts, bitmap);

  dim3 gS(NSCAN_BLOCKS, B);
  k_block_reduce<<<gS, T, 0, stream>>>(bitmap, blockSums);
  k_scan_blocksums<<<B, T, 0, stream>>>(blockSums);
  k_scan_emit<<<gS, T, 0, stream>>>(bitmap, blockSums, prefix, outVC, outMask);

  k_scatter<<<gP, T, 0, stream>>>(pts, bitmap, prefix, slotIdx);

  size_t totVox = (size_t)B * MAX_VOX * MAX_PTS;
  k_write_voxels<<<(int)((totVox + T - 1) / T), T, 0, stream>>>(pts, slotIdx, outVox, B);
}